// NolinerSEM_5420248727806
// MI455X (gfx1250) — compile-verified
//
#include <hip/hip_runtime.h>
#include <hip/hip_bf16.h>

// ---------------------------------------------------------------------------
// The reference network output is provably independent of x:
//   stride-2 conv with kernel width 1 and W-padding 1 on a W=1 input samples
//   only the zero-pad columns, so layer-1 output == relu(b1) everywhere.
// The whole pipeline collapses to 90 tiny per-network evaluations whose
// result is broadcast across the 2048-row batch (737 KB ~ 32 ns at 23.3 TB/s;
// launch-latency bound).
//
// Kernel 1: one wave32 per network. The two small matmuls
//   (3x48 @ 48x32) and (3x96 @ 96x64)
// run on V_WMMA_F32_16X16X4_F32 (f32 WMMA, exact precision). The kh-masked
// A matrices are materialized zero-padded in LDS up front so the WMMA loops
// are branch-free ds_load_b64 + global_load_b64 + wmma streams (EXEC all-1s).
// Kernel 2: broadcast the 90-vector to the 2048x90 output.
// ---------------------------------------------------------------------------

typedef __attribute__((ext_vector_type(2))) float v2f;
typedef __attribute__((ext_vector_type(8))) float v8f;

#define NNET 90
#define BATCH 2048

__global__ __launch_bounds__(32)
void net_eval_kernel(const float* __restrict__ b1,  // (90,16)
                     const float* __restrict__ w2,  // (90,32,16,3,1)
                     const float* __restrict__ b2,  // (90,32)
                     const float* __restrict__ w3,  // (90,64,32,3,1)
                     const float* __restrict__ b3,  // (90,64)
                     const float* __restrict__ fw,  // (90,1,64)
                     const float* __restrict__ fb,  // (90,1)
                     float* __restrict__ out_vec)   // (90)
{
    const int n    = blockIdx.x;        // network id
    const int lane = threadIdx.x;       // 0..31 (wave32)
    const int half = lane >> 4;         // lane 0-15 vs 16-31
    const int m    = lane & 15;         // WMMA row (A) / column (B,C,D)

    __shared__ float s_h1[16];
    __shared__ float s_A2[16 * 48];     // layer-2 A matrix, rows 3..15 zero
    __shared__ float s_A3[16 * 96];     // layer-3 A matrix, rows 3..15 zero
    __shared__ float s_e0[32];
    __shared__ float s_m [32];
    __shared__ float s_e1[32];
    __shared__ float s_p [64];

    // ---- setup: h1 = relu(b1) (layer-1 conv samples only zero padding);
    //      zero the A matrices once so rows 3..15 contribute nothing.
    if (lane < 16) s_h1[lane] = fmaxf(b1[n * 16 + lane], 0.0f);
    for (int i = lane; i < 16 * 48; i += 32) s_A2[i] = 0.0f;
    for (int i = lane; i < 16 * 96; i += 32) s_A3[i] = 0.0f;
    __syncthreads();

    // ---- build A2 rows 0..2 over K = 3k+kh (k=0..15):
    //   row0 = e0 (taps kh=1,2), row1 = m (kh=0,1,2), row2 = e1 (kh=0,1)
    for (int i = lane; i < 3 * 48; i += 32) {
        int row = i / 48;
        int r   = i - row * 48;
        int k   = r / 3;
        int kh  = r - 3 * k;
        bool on = (row == 1) || (row == 0 && kh != 0) || (row == 2 && kh != 2);
        s_A2[row * 48 + r] = on ? s_h1[k] : 0.0f;
    }
    __syncthreads();

    // ---- layer 2: D(3x32) = A2(3x48) * B(48x32) + b2
    // B[K, c] = w2[n, c, k, kh] = w2_flat[n*1536 + c*48 + K]
    const float* w2g = w2 + (size_t)n * 32 * 16 * 3;
    const float* b2g = b2 + n * 32;

    #pragma unroll
    for (int tile = 0; tile < 2; ++tile) {
        const int   c    = m + 16 * tile;
        const float bias = b2g[c];
        v8f acc = {bias, bias, bias, bias, bias, bias, bias, bias};
        // f32 16x16x4 fragment K map: VGPR0 -> K = 4s + 2*half, VGPR1 -> K+1
        const float* ar = s_A2 + m * 48 + 2 * half;
        const float* br = w2g + c * 48 + 2 * half;
        #pragma unroll
        for (int s = 0; s < 12; ++s) {
            v2f a, b;
            a.x = ar[4 * s];
            a.y = ar[4 * s + 1];
            b.x = br[4 * s];
            b.y = br[4 * s + 1];
            acc = __builtin_amdgcn_wmma_f32_16x16x4_f32(
                false, a, false, b, (short)0, acc, false, false);
        }
        if (lane < 16) {  // C/D layout: VGPR v holds row M=v for lanes 0-15
            s_e0[c] = fmaxf(acc[0], 0.0f);
            s_m [c] = fmaxf(acc[1], 0.0f);
            s_e1[c] = fmaxf(acc[2], 0.0f);
        }
    }
    __syncthreads();

    // ---- build A3 rows 0..2 over K = 3k+kh (k=0..31):
    //   row0: kh1->e0, kh2->m ; row1: all m ; row2: kh0->m, kh1->e1
    for (int i = lane; i < 3 * 96; i += 32) {
        int row = i / 96;
        int r   = i - row * 96;
        int k   = r / 3;
        int kh  = r - 3 * k;
        float v;
        if (row == 1)      v = s_m[k];
        else if (row == 0) v = (kh == 1) ? s_e0[k] : ((kh == 2) ? s_m[k] : 0.0f);
        else               v = (kh == 0) ? s_m[k]  : ((kh == 1) ? s_e1[k] : 0.0f);
        s_A3[row * 96 + r] = v;
    }
    __syncthreads();

    // ---- layer 3: D(3x64) = A3(3x96) * B(96x64) + b3, then pooled p[c]
    // B[K, c] = w3[n, c, k, kh] = w3_flat[n*6144 + c*96 + K]
    const float* w3g = w3 + (size_t)n * 64 * 32 * 3;
    const float* b3g = b3 + n * 64;

    #pragma unroll
    for (int tile = 0; tile < 4; ++tile) {
        const int   c    = m + 16 * tile;
        const float bias = b3g[c];
        v8f acc = {bias, bias, bias, bias, bias, bias, bias, bias};
        const float* ar = s_A3 + m * 96 + 2 * half;
        const float* br = w3g + c * 96 + 2 * half;
        #pragma unroll
        for (int s = 0; s < 24; ++s) {
            v2f a, b;
            a.x = ar[4 * s];
            a.y = ar[4 * s + 1];
            b.x = br[4 * s];
            b.y = br[4 * s + 1];
            acc = __builtin_amdgcn_wmma_f32_16x16x4_f32(
                false, a, false, b, (short)0, acc, false, false);
        }
        if (lane < 16) {
            float h3_0   = fmaxf(acc[0], 0.0f);
            float h3_mid = fmaxf(acc[1], 0.0f);
            float h3_11  = fmaxf(acc[2], 0.0f);
            float pb     = fmaxf(b3g[c], 0.0f);   // w=0 column: 12 x relu(b3)
            s_p[c] = (12.0f * pb + h3_0 + 10.0f * h3_mid + h3_11)
                     * (1.0f / 24.0f);
        }
    }
    __syncthreads();

    // ---- head: out[n] = dot(fw[n], p) + fb[n], wave32 reduction ----
    const float* fwg = fw + n * 64;
    float partial = fwg[lane]      * s_p[lane]
                  + fwg[lane + 32] * s_p[lane + 32];
    #pragma unroll
    for (int off = 16; off > 0; off >>= 1)
        partial += __shfl_xor(partial, off, 32);
    if (lane == 0) out_vec[n] = partial + fb[n];
}

__global__ void broadcast_kernel(const float* __restrict__ vec,
                                 float* __restrict__ out, int total) {
    int i = blockIdx.x * blockDim.x + threadIdx.x;
    if (i < total) out[i] = vec[i % NNET];
}

extern "C" void kernel_launch(void* const* d_in, const int* in_sizes, int n_in,
                              void* d_out, int out_size, void* d_ws, size_t ws_size,
                              hipStream_t stream) {
    (void)in_sizes; (void)n_in; (void)out_size; (void)ws_size;
    // setup_inputs order: x, w1, b1, w2, b2, w3, b3, fw, fb
    // x (d_in[0]) and w1 (d_in[1]) provably do not affect the output.
    const float* b1 = (const float*)d_in[2];
    const float* w2 = (const float*)d_in[3];
    const float* b2 = (const float*)d_in[4];
    const float* w3 = (const float*)d_in[5];
    const float* b3 = (const float*)d_in[6];
    const float* fw = (const float*)d_in[7];
    const float* fb = (const float*)d_in[8];
    float* out = (float*)d_out;
    float* vec = (float*)d_ws;   // 90 floats of scratch

    net_eval_kernel<<<NNET, 32, 0, stream>>>(b1, w2, b2, w3, b3, fw, fb, vec);

    const int total = BATCH * NNET;
    broadcast_kernel<<<(total + 255) / 256, 256, 0, stream>>>(vec, out, total);
}